// Model_65678639890859
// MI455X (gfx1250) — compile-verified
//
#include <hip/hip_runtime.h>
#include <hip/hip_bf16.h>

#define V 6
#define U 4
#define SEQLEN 64

typedef __attribute__((ext_vector_type(2))) float v2f;
typedef __attribute__((ext_vector_type(8))) float v8f;

__device__ __forceinline__ float sigmoid_f(float x) {
    return 1.0f / (1.0f + __expf(-x));
}

// ---------------------------------------------------------------------------
// Phase 1: tiny serial LSTM scan on ONE wave32.
// Lane l = j*4+u (24 active lanes) owns gate element (j,u).
// h[j][0..3] gathered with 4 shuffles; linear-output reduction is a 4-lane
// xor-butterfly. 64 fully serial steps, each a short dependency chain.
// ---------------------------------------------------------------------------
__global__ __launch_bounds__(32)
void lstm_scan_kernel(const float* __restrict__ vel,
                      const float* __restrict__ Wix, const float* __restrict__ Wih, const float* __restrict__ bi,
                      const float* __restrict__ Wfx, const float* __restrict__ Wfh, const float* __restrict__ bf,
                      const float* __restrict__ Wox, const float* __restrict__ Woh, const float* __restrict__ bo,
                      const float* __restrict__ Wgx, const float* __restrict__ Wgh, const float* __restrict__ bg,
                      const float* __restrict__ linear, const float* __restrict__ bl,
                      const float* __restrict__ h0, const float* __restrict__ c0,
                      float* __restrict__ vels_out) {
    const int lane = threadIdx.x;        // 0..31
    const bool act = lane < V * U;       // 24 active lanes
    const int j = lane >> 2;

    const float wix = act ? Wix[lane] : 0.f, bi_ = act ? bi[lane] : 0.f;
    const float wfx = act ? Wfx[lane] : 0.f, bf_ = act ? bf[lane] : 0.f;
    const float wox = act ? Wox[lane] : 0.f, bo_ = act ? bo[lane] : 0.f;
    const float wgx = act ? Wgx[lane] : 0.f, bg_ = act ? bg[lane] : 0.f;
    float wih[4], wfh[4], woh[4], wgh[4];
#pragma unroll
    for (int k = 0; k < 4; ++k) {
        wih[k] = act ? Wih[lane * 4 + k] : 0.f;
        wfh[k] = act ? Wfh[lane * 4 + k] : 0.f;
        woh[k] = act ? Woh[lane * 4 + k] : 0.f;
        wgh[k] = act ? Wgh[lane * 4 + k] : 0.f;
    }
    const float lin = act ? linear[lane] : 0.f;
    const float blj = act ? bl[j] : 0.f;

    float h = act ? h0[lane] : 0.f;
    float c = act ? c0[lane] : 0.f;
    float x = act ? vel[j] : 0.f;   // step input x[j] (same for all u in group j)

    auto step = [&]() {
        const int base = j << 2;
        float hj0 = __shfl(h, base + 0, 32);
        float hj1 = __shfl(h, base + 1, 32);
        float hj2 = __shfl(h, base + 2, 32);
        float hj3 = __shfl(h, base + 3, 32);
        float gi = wih[0]*hj0 + wih[1]*hj1 + wih[2]*hj2 + wih[3]*hj3;
        float gf = wfh[0]*hj0 + wfh[1]*hj1 + wfh[2]*hj2 + wfh[3]*hj3;
        float go = woh[0]*hj0 + woh[1]*hj1 + woh[2]*hj2 + woh[3]*hj3;
        float gg = wgh[0]*hj0 + wgh[1]*hj1 + wgh[2]*hj2 + wgh[3]*hj3;
        float it = sigmoid_f(wix * x + gi + bi_);
        float ft = sigmoid_f(wfx * x + gf + bf_);
        float ot = sigmoid_f(wox * x + go + bo_);
        float g  = sigmoid_f(wgx * x + gg + bg_);   // source uses sigmoid here
        c = ft * c + it * g;
        h = ot * sigmoid_f(c);                      // source uses sigmoid, not tanh
        float t = lin * h;
        t += __shfl_xor(t, 1, 32);
        t += __shfl_xor(t, 2, 32);                  // all 4 lanes of group j hold sum
        x = tanhf(t + blj);                         // out[j] becomes next input
    };

    step();              // x = out_0
    float vacc = x;      // vels = out_0 (i == 0 branch)
    for (int i = 1; i < SEQLEN; ++i) {
        vacc += x;       // vels += out_{i-1}
        step();          // x = out_i
    }

    if (act && (lane & 3) == 0) vels_out[j] = vacc;
}

// ---------------------------------------------------------------------------
// Phase 2: bandwidth-bound streaming matvec via V_WMMA_F32_16X16X4_F32.
// A (16x4): vels[k] replicated over M  -> lanes 0-15 hold {v0,v1}, 16-31 {v2,v3}
// B (4x16): 16 rows of Ls             -> lane n loads row n k={0,1} (float2),
//                                        lane n+16 loads row n k={2,3}
// Second WMMA covers k={4,5}; its A hi-half is {0,0}, so B's hi-lane contents
// are multiplied by zero -> load row+4 unconditionally (lanes n and n+16 read
// the same 8 bytes, coalesced) and avoid any EXEC manipulation around WMMA.
// Two 16-row tiles per iteration: lo lanes carry tile0 results, hi lanes carry
// tile1 results, so the store is a fully coalesced float2 per lane at row
// t*32+lane. Non-temporal hints: single-pass stream, no L2 reuse.
// ---------------------------------------------------------------------------
__global__ __launch_bounds__(256)
void flow_matvec_wmma_kernel(const float* __restrict__ Lsx,
                             const float* __restrict__ Lsy,
                             const float* __restrict__ vels,
                             float* __restrict__ out,
                             int N, int numT32) {
    const int lane = threadIdx.x & 31;
    const int n = lane & 15;
    const bool hi = lane >= 16;

    const float v0 = vels[0], v1 = vels[1], v2 = vels[2];
    const float v3 = vels[3], v4 = vels[4], v5 = vels[5];

    v2f a0, a1;
    a0[0] = hi ? v2 : v0;
    a0[1] = hi ? v3 : v1;
    a1[0] = hi ? 0.f : v4;
    a1[1] = hi ? 0.f : v5;

    const int wave   = (int)((blockIdx.x * blockDim.x + threadIdx.x) >> 5);
    const int nwaves = (int)((gridDim.x * blockDim.x) >> 5);
    const int koff   = hi ? 2 : 0;

    for (int t = wave; t < numT32; t += nwaves) {     // wave-uniform loop
        long rA = (long)t * 32 + n;                   // tile0 rows
        long rB = rA + 16;                            // tile1 rows
        long ra = rA < (long)N ? rA : (long)N - 1;    // branchless tail clamp
        long rb = rB < (long)N ? rB : (long)N - 1;
        const float* pxa = Lsx + ra * 6;
        const float* pya = Lsy + ra * 6;
        const float* pxb = Lsx + rb * 6;
        const float* pyb = Lsy + rb * 6;

        // 8 independent b64 loads in flight before any wait
        v2f bx0a = __builtin_nontemporal_load((const v2f*)(pxa + koff));
        v2f bx1a = __builtin_nontemporal_load((const v2f*)(pxa + 4));
        v2f by0a = __builtin_nontemporal_load((const v2f*)(pya + koff));
        v2f by1a = __builtin_nontemporal_load((const v2f*)(pya + 4));
        v2f bx0b = __builtin_nontemporal_load((const v2f*)(pxb + koff));
        v2f bx1b = __builtin_nontemporal_load((const v2f*)(pxb + 4));
        v2f by0b = __builtin_nontemporal_load((const v2f*)(pyb + koff));
        v2f by1b = __builtin_nontemporal_load((const v2f*)(pyb + 4));

        v8f dxa = {}, dya = {}, dxb = {}, dyb = {};
        dxa = __builtin_amdgcn_wmma_f32_16x16x4_f32(false, a0, false, bx0a, (short)0, dxa, false, false);
        dxa = __builtin_amdgcn_wmma_f32_16x16x4_f32(false, a1, false, bx1a, (short)0, dxa, false, false);
        dya = __builtin_amdgcn_wmma_f32_16x16x4_f32(false, a0, false, by0a, (short)0, dya, false, false);
        dya = __builtin_amdgcn_wmma_f32_16x16x4_f32(false, a1, false, by1a, (short)0, dya, false, false);
        dxb = __builtin_amdgcn_wmma_f32_16x16x4_f32(false, a0, false, bx0b, (short)0, dxb, false, false);
        dxb = __builtin_amdgcn_wmma_f32_16x16x4_f32(false, a1, false, bx1b, (short)0, dxb, false, false);
        dyb = __builtin_amdgcn_wmma_f32_16x16x4_f32(false, a0, false, by0b, (short)0, dyb, false, false);
        dyb = __builtin_amdgcn_wmma_f32_16x16x4_f32(false, a1, false, by1b, (short)0, dyb, false, false);

        // lane n (lo)    -> tile0 row t*32+n      : result in dxa[0]/dya[0]
        // lane 16+n (hi) -> tile1 row t*32+16+n   : result in dxb[0]/dyb[0]
        // => every lane stores row t*32+lane: fully coalesced 256B wave store
        long rowS = (long)t * 32 + lane;
        v2f o;
        o[0] = hi ? dxb[0] : dxa[0];
        o[1] = hi ? dyb[0] : dya[0];
        if (rowS < (long)N) {
            __builtin_nontemporal_store(o, (v2f*)(out + rowS * 2));
        }
    }
}

extern "C" void kernel_launch(void* const* d_in, const int* in_sizes, int n_in,
                              void* d_out, int out_size, void* d_ws, size_t ws_size,
                              hipStream_t stream) {
    const float* vel = (const float*)d_in[0];
    const float* Lsx = (const float*)d_in[1];
    const float* Lsy = (const float*)d_in[2];
    const float* Wix = (const float*)d_in[3];
    const float* Wih = (const float*)d_in[4];
    const float* bi  = (const float*)d_in[5];
    const float* Wfx = (const float*)d_in[6];
    const float* Wfh = (const float*)d_in[7];
    const float* bf  = (const float*)d_in[8];
    const float* Wox = (const float*)d_in[9];
    const float* Woh = (const float*)d_in[10];
    const float* bo  = (const float*)d_in[11];
    const float* Wgx = (const float*)d_in[12];
    const float* Wgh = (const float*)d_in[13];
    const float* bg  = (const float*)d_in[14];
    const float* lin = (const float*)d_in[15];
    const float* bl  = (const float*)d_in[16];
    const float* h0  = (const float*)d_in[17];
    const float* c0  = (const float*)d_in[18];

    float* vels_ws = (float*)d_ws;   // 6 floats of scratch
    const int N = in_sizes[1] / V;

    lstm_scan_kernel<<<1, 32, 0, stream>>>(vel, Wix, Wih, bi, Wfx, Wfh, bf,
                                           Wox, Woh, bo, Wgx, Wgh, bg,
                                           lin, bl, h0, c0, vels_ws);

    const int numT32 = (N + 31) / 32;    // 32-row tile pairs
    int blocks = (numT32 + 7) / 8;       // 8 waves per 256-thread block
    if (blocks > 4096) blocks = 4096;    // persistent waves, grid-stride
    flow_matvec_wmma_kernel<<<blocks, 256, 0, stream>>>(Lsx, Lsy, vels_ws,
                                                        (float*)d_out, N, numT32);
}